// HebbianLearning_41111426957364
// MI455X (gfx1250) — compile-verified
//
#include <hip/hip_runtime.h>
#include <hip/hip_bf16.h>
#include <math.h>

// Problem dims (fixed by reference): B=2048, C=32, K=7 -> D=1568, O=512
#define B_N 2048
#define O_N 512
#define D_N 1568

typedef float v2f __attribute__((ext_vector_type(2)));
typedef float v8f __attribute__((ext_vector_type(8)));

// ---------------------------------------------------------------------------
// Wave-level 16x16 fp32 WMMA tile GEMM (V_WMMA_F32_16X16X4_F32, wave32).
// VGPR layout assumptions (CDNA5 ISA 7.12.2):
//   A 16x4 : lane L -> m = L&15 ; VGPR j holds K = (L>>4)*2 + j
//   B 4x16 : lane L -> n = L&15 ; VGPR j holds K = (L>>4)*2 + j
//   C/D    : lane L -> n = L&15 ; VGPR i holds m = (L>>4)*8 + i
// ---------------------------------------------------------------------------
template <typename FA, typename FB, typename FS>
__device__ __forceinline__ void wmma_tile_f32(int K, FA loadA, FB loadB, FS store) {
  const int lane = threadIdx.x & 31;
  const int l16  = lane & 15;
  const int koff = (lane >> 4) * 2;
  v8f acc = {};
  for (int k = 0; k < K; k += 4) {
    v2f a, b;
    a.x = loadA(l16, k + koff);
    a.y = loadA(l16, k + koff + 1);
    b.x = loadB(k + koff,     l16);
    b.y = loadB(k + koff + 1, l16);
    acc = __builtin_amdgcn_wmma_f32_16x16x4_f32(false, a, false, b, (short)0, acc,
                                                false, false);
  }
  const int mbase = (lane >> 4) * 8;
#pragma unroll
  for (int i = 0; i < 8; ++i) store(mbase + i, l16, acc[i]);
}

// ---------------------------------------------------------------------------
// CDNA5 async global->LDS helpers (ASYNCcnt path, CDNA5 ISA)
// ---------------------------------------------------------------------------
__device__ __forceinline__ unsigned lds_addr(const void* p) {
  // generic -> LDS addrspace(3) pointer -> 32-bit LDS byte offset
  return (unsigned)(unsigned long long)(const __attribute__((address_space(3))) void*)p;
}

__device__ __forceinline__ void async_ld_b128(unsigned lds_off, const float* g) {
  unsigned long long ga = (unsigned long long)g;
  asm volatile("global_load_async_to_lds_b128 %0, %1, off"
               :: "v"(lds_off), "v"(ga) : "memory");
}

__device__ __forceinline__ void wait_async0() {
  asm volatile("s_wait_asynccnt 0" ::: "memory");
}

// ---------------------------------------------------------------------------
// S = X * W^T   (B x O), K = D.
// Block-tiled: 64x64 output block, K-chunks of 16 staged in LDS via
// double-buffered GLOBAL_LOAD_ASYNC_TO_LDS_B128.  8 waves as 4(M)x2(N),
// each wave computes 16x32 (2 accumulators, A fragment reused).
// ---------------------------------------------------------------------------
#define BM 64
#define BN 64
#define KC 16
#define LDSTR 20                 // LDS row stride in floats (16 + 4 pad, 16B-aligned)
#define TILE_F (64 * LDSTR)      // 1280 floats per staged tile
#define BUF_F  (2 * TILE_F)      // Xs + Ws per buffer

__global__ void k_gemm_S(const float* __restrict__ X, const float* __restrict__ W,
                         float* __restrict__ S) {
  __shared__ __align__(16) float smem[2 * BUF_F];   // ~20 KB of the 320 KB WGP LDS
  const int TNB = O_N / BN;                         // 8
  const int tm = blockIdx.x / TNB, tn = blockIdx.x % TNB;
  const int m0 = tm * BM, n0 = tn * BN;

  const int t = threadIdx.x;
  const int wave = t >> 5, lane = t & 31;
  const int wm = wave >> 1, wn = wave & 1;          // wave tile: rows wm*16, cols wn*32
  const int l16 = lane & 15, koff = (lane >> 4) * 2;

  // Staging assignment: thread -> one float4 of X-chunk and one of W-chunk.
  const int lrow = t >> 2;                          // 0..63
  const int lkg  = (t & 3) * 4;                     // 0,4,8,12
  const unsigned x_lds0 = lds_addr(smem) + (unsigned)(lrow * LDSTR + lkg) * 4u;
  const unsigned w_lds0 = x_lds0 + (unsigned)TILE_F * 4u;
  const float* xg = X + (size_t)(m0 + lrow) * D_N + lkg;
  const float* wg = W + (size_t)(n0 + lrow) * D_N + lkg;

  // Prologue: chunk 0 -> buffer 0.
  async_ld_b128(x_lds0, xg);
  async_ld_b128(w_lds0, wg);

  v8f acc0 = {}, acc1 = {};
  const int NCH = D_N / KC;                         // 98
  for (int c = 0; c < NCH; ++c) {
    wait_async0();        // my staged chunk(s) are in LDS
    __syncthreads();      // everyone's are; all waves done reading other buffer
    const int buf = c & 1;
    if (c + 1 < NCH) {    // kick off next chunk's DMA into the other buffer
      const unsigned bo = (unsigned)(((c + 1) & 1) * BUF_F) * 4u;
      async_ld_b128(x_lds0 + bo, xg + (size_t)(c + 1) * KC);
      async_ld_b128(w_lds0 + bo, wg + (size_t)(c + 1) * KC);
    }
    const float* Xs = smem + buf * BUF_F;
    const float* Ws = Xs + TILE_F;
#pragma unroll
    for (int ks = 0; ks < KC; ks += 4) {
      v2f a, b0, b1;
      const int ar = (wm * 16 + l16) * LDSTR + ks + koff;
      a.x = Xs[ar];
      a.y = Xs[ar + 1];
      const int br0 = (wn * 32 + l16) * LDSTR + ks + koff;
      b0.x = Ws[br0];
      b0.y = Ws[br0 + 1];
      const int br1 = (wn * 32 + 16 + l16) * LDSTR + ks + koff;
      b1.x = Ws[br1];
      b1.y = Ws[br1 + 1];
      acc0 = __builtin_amdgcn_wmma_f32_16x16x4_f32(false, a, false, b0, (short)0, acc0,
                                                   false, false);
      acc1 = __builtin_amdgcn_wmma_f32_16x16x4_f32(false, a, false, b1, (short)0, acc1,
                                                   false, false);
    }
  }

  const int mb = (lane >> 4) * 8;
#pragma unroll
  for (int i = 0; i < 8; ++i) {
    const size_t row = (size_t)(m0 + wm * 16 + mb + i) * O_N + n0 + wn * 32;
    S[row + l16]      = acc0[i];
    S[row + 16 + l16] = acc1[i];
  }
}

// ---------------------------------------------------------------------------
// Per-sample logits / ff_loss / coef:  l_b = sum_o s^2 - O
// coef_b = 2*(sigmoid(l_b) - label_b)/B ;  ff_loss += logistic(l_b)/B
// ---------------------------------------------------------------------------
__global__ void k_logits(const float* __restrict__ S, const float* __restrict__ label,
                         float* __restrict__ coef, float* __restrict__ ff_loss) {
  __shared__ float red[256];
  const int b = blockIdx.x;
  const float* row = S + (size_t)b * O_N;
  const float s0 = row[threadIdx.x];
  const float s1 = row[threadIdx.x + 256];
  red[threadIdx.x] = s0 * s0 + s1 * s1;
  __syncthreads();
  for (int off = 128; off > 0; off >>= 1) {
    if (threadIdx.x < off) red[threadIdx.x] += red[threadIdx.x + off];
    __syncthreads();
  }
  if (threadIdx.x == 0) {
    const float l  = red[0] - (float)O_N;
    const float lb = label[b];
    const float sig = 1.0f / (1.0f + expf(-l));
    coef[b] = 2.0f * (sig - lb) / (float)B_N;
    const float loss = fmaxf(l, 0.0f) - l * lb + log1pf(expf(-fabsf(l)));
    atomicAdd(ff_loss, loss / (float)B_N);
  }
}

// ---------------------------------------------------------------------------
// r_sum[o] = sum_b |label_b * s[b,o]| ; rinv = 1/(r_sum + (r_sum==0))
// ---------------------------------------------------------------------------
__global__ void k_rsum(const float* __restrict__ S, const float* __restrict__ label,
                       float* __restrict__ rinv) {
  __shared__ float red[256];
  const int o = blockIdx.x;
  float acc = 0.0f;
  for (int b = threadIdx.x; b < B_N; b += 256)
    acc += fabsf(S[(size_t)b * O_N + o] * label[b]);
  red[threadIdx.x] = acc;
  __syncthreads();
  for (int off = 128; off > 0; off >>= 1) {
    if (threadIdx.x < off) red[threadIdx.x] += red[threadIdx.x + off];
    __syncthreads();
  }
  if (threadIdx.x == 0) {
    float rs = red[0];
    if (rs == 0.0f) rs = 1.0f;
    rinv[o] = 1.0f / rs;
  }
}

// ---------------------------------------------------------------------------
// Fused:  ff_grad = (coef .* S)^T X   and   P = U^T X,  u[b,o]=label_b*s*|s|
// (O x D) each, K = B.  Two accumulators share the X B-operand fragment.
// ---------------------------------------------------------------------------
__global__ void k_gemm_ffgrad_P(const float* __restrict__ S, const float* __restrict__ X,
                                const float* __restrict__ coef,
                                const float* __restrict__ label,
                                float* __restrict__ ffgrad, float* __restrict__ P) {
  const int wave = (blockIdx.x * blockDim.x + threadIdx.x) >> 5;
  const int TN = D_N / 16;
  const int tm = wave / TN, tn = wave % TN;
  const int o0 = tm * 16, d0 = tn * 16;
  const int lane = threadIdx.x & 31;
  const int l16 = lane & 15;
  const int koff = (lane >> 4) * 2;
  v8f accF = {}, accP = {};
  for (int b = 0; b < B_N; b += 4) {
    const int b0 = b + koff, b1 = b0 + 1;
    const float s0 = S[(size_t)b0 * O_N + o0 + l16];
    const float s1 = S[(size_t)b1 * O_N + o0 + l16];
    v2f af, au, bx;
    af.x = coef[b0] * s0;
    af.y = coef[b1] * s1;
    au.x = label[b0] * s0 * fabsf(s0);
    au.y = label[b1] * s1 * fabsf(s1);
    bx.x = X[(size_t)b0 * D_N + d0 + l16];
    bx.y = X[(size_t)b1 * D_N + d0 + l16];
    accF = __builtin_amdgcn_wmma_f32_16x16x4_f32(false, af, false, bx, (short)0, accF,
                                                 false, false);
    accP = __builtin_amdgcn_wmma_f32_16x16x4_f32(false, au, false, bx, (short)0, accP,
                                                 false, false);
  }
  const int mb = (lane >> 4) * 8;
#pragma unroll
  for (int i = 0; i < 8; ++i) {
    const size_t idx = (size_t)(o0 + mb + i) * D_N + d0 + l16;
    ffgrad[idx] = accF[i];
    P[idx]      = accP[i];
  }
}

// ---------------------------------------------------------------------------
// Mtri = tril(U^T R)   (O x O), K = B;  r[b,o] = label_b * s[b,o]
// ---------------------------------------------------------------------------
__global__ void k_gemm_M(const float* __restrict__ S, const float* __restrict__ label,
                         float* __restrict__ Mtri) {
  const int wave = (blockIdx.x * blockDim.x + threadIdx.x) >> 5;
  const int TN = O_N / 16;
  const int tm = wave / TN, tn = wave % TN;
  const int o0 = tm * 16, p0 = tn * 16;
  wmma_tile_f32(B_N,
      [&](int m, int k) {  // u[k, o0+m]
        const float s = S[(size_t)k * O_N + o0 + m];
        return label[k] * s * fabsf(s);
      },
      [&](int k, int n) {  // r[k, p0+n]
        return label[k] * S[(size_t)k * O_N + p0 + n];
      },
      [&](int m, int n, float v) {
        const int o = o0 + m, p = p0 + n;
        Mtri[(size_t)o * O_N + p] = (p <= o) ? v : 0.0f;  // cumsum: o' <= o
      });
}

// ---------------------------------------------------------------------------
// Gc = (W W^T) .* (O - max(o,p))   (O x O), K = D
// ---------------------------------------------------------------------------
__global__ void k_gemm_G(const float* __restrict__ W, float* __restrict__ Gc) {
  const int wave = (blockIdx.x * blockDim.x + threadIdx.x) >> 5;
  const int TN = O_N / 16;
  const int tm = wave / TN, tn = wave % TN;
  const int o0 = tm * 16, p0 = tn * 16;
  wmma_tile_f32(D_N,
      [&](int m, int k) { return W[(size_t)(o0 + m) * D_N + k]; },
      [&](int k, int n) { return W[(size_t)(p0 + n) * D_N + k]; },
      [&](int m, int n, float v) {
        const int o = o0 + m, p = p0 + n;
        const int mx = (o > p) ? o : p;
        Gc[(size_t)o * O_N + p] = v * (float)(O_N - mx);
      });
}

// ---------------------------------------------------------------------------
// T = R * Gc   (B x O), K = O
// ---------------------------------------------------------------------------
__global__ void k_gemm_T(const float* __restrict__ S, const float* __restrict__ label,
                         const float* __restrict__ Gc, float* __restrict__ T) {
  const int wave = (blockIdx.x * blockDim.x + threadIdx.x) >> 5;
  const int TN = O_N / 16;
  const int tm = wave / TN, tn = wave % TN;
  const int b0 = tm * 16, p0 = tn * 16;
  wmma_tile_f32(O_N,
      [&](int m, int k) { return label[b0 + m] * S[(size_t)(b0 + m) * O_N + k]; },
      [&](int k, int n) { return Gc[(size_t)k * O_N + p0 + n]; },
      [&](int m, int n, float v) { T[(size_t)(b0 + m) * O_N + p0 + n] = v; });
}

// ---------------------------------------------------------------------------
// recon = sum_b label_b * ( O*||x_b||^2 - 2*sum_o (O-o) s^2 + sum_o s*T[b,o] )
// (labels are {0,1} so label powers collapse)
// ---------------------------------------------------------------------------
__global__ void k_recon(const float* __restrict__ S, const float* __restrict__ X,
                        const float* __restrict__ T, const float* __restrict__ label,
                        float* __restrict__ recon) {
  __shared__ float red[256];
  const int b = blockIdx.x;
  const int t = threadIdx.x;
  float xn = 0.0f;
  for (int d = t; d < D_N; d += 256) {
    const float x = X[(size_t)b * D_N + d];
    xn += x * x;
  }
  float pb = 0.0f, pq = 0.0f;
  for (int o = t; o < O_N; o += 256) {
    const float s = S[(size_t)b * O_N + o];
    pb += (float)(O_N - o) * s * s;
    pq += s * T[(size_t)b * O_N + o];
  }
  red[t] = (float)O_N * xn - 2.0f * pb + pq;
  __syncthreads();
  for (int off = 128; off > 0; off >>= 1) {
    if (t < off) red[t] += red[t + off];
    __syncthreads();
  }
  if (t == 0) {
    const float lb = label[b];
    if (lb != 0.0f) atomicAdd(recon, lb * red[0]);
  }
}

// ---------------------------------------------------------------------------
// dw = (P - Mtri*W) * rinv[o]   (O x D), K = O
// ---------------------------------------------------------------------------
__global__ void k_gemm_dw(const float* __restrict__ Mtri, const float* __restrict__ W,
                          const float* __restrict__ P, const float* __restrict__ rinv,
                          float* __restrict__ dw) {
  const int wave = (blockIdx.x * blockDim.x + threadIdx.x) >> 5;
  const int TN = D_N / 16;
  const int tm = wave / TN, tn = wave % TN;
  const int o0 = tm * 16, d0 = tn * 16;
  const int l16 = threadIdx.x & 15;
  __builtin_prefetch(&Mtri[(size_t)(o0 + l16) * O_N], 0, 1);  // global_prefetch_b8
  wmma_tile_f32(O_N,
      [&](int m, int k) { return Mtri[(size_t)(o0 + m) * O_N + k]; },
      [&](int k, int n) { return W[(size_t)k * D_N + d0 + n]; },
      [&](int m, int n, float v) {
        const size_t idx = (size_t)(o0 + m) * D_N + d0 + n;
        dw[idx] = (P[idx] - v) * rinv[o0 + m];
      });
}

__global__ void k_init(float* __restrict__ recon, float* __restrict__ ffl) {
  if (threadIdx.x == 0) {
    *recon = 0.0f;
    *ffl = 0.0f;
  }
}

// ---------------------------------------------------------------------------
// Launch.  d_in order: data (B,C,7,7)=B x D, label (B), weight (O,C,7,7)=O x D,
// epoch (unused).  d_out: [S | dw | recon | ff_loss | ff_grad].
// ---------------------------------------------------------------------------
extern "C" void kernel_launch(void* const* d_in, const int* in_sizes, int n_in,
                              void* d_out, int out_size, void* d_ws, size_t ws_size,
                              hipStream_t stream) {
  const float* X     = (const float*)d_in[0];
  const float* label = (const float*)d_in[1];
  const float* W     = (const float*)d_in[2];

  float* out   = (float*)d_out;
  float* S     = out;                                  // B*O
  float* dw    = S + (size_t)B_N * O_N;                // O*D
  float* recon = dw + (size_t)O_N * D_N;               // 1
  float* ffl   = recon + 1;                            // 1
  float* ffgr  = ffl + 1;                              // O*D

  float* ws   = (float*)d_ws;
  float* coef = ws;                                    // B
  float* rinv = coef + B_N;                            // O
  float* P    = rinv + O_N;                            // O*D
  float* Gc   = P + (size_t)O_N * D_N;                 // O*O
  float* Mtri = Gc + (size_t)O_N * O_N;                // O*O
  float* T    = Mtri + (size_t)O_N * O_N;              // B*O   (~9.5 MB total)

  const int TPB = 256;  // 8 wave32 waves
  k_init<<<1, 32, 0, stream>>>(recon, ffl);
  k_gemm_S<<<(B_N / BM) * (O_N / BN), TPB, 0, stream>>>(X, W, S);
  k_logits<<<B_N, TPB, 0, stream>>>(S, label, coef, ffl);
  k_rsum<<<O_N, TPB, 0, stream>>>(S, label, rinv);
  k_gemm_ffgrad_P<<<(O_N / 16) * (D_N / 16) / 8, TPB, 0, stream>>>(S, X, coef, label,
                                                                   ffgr, P);
  k_gemm_M<<<(O_N / 16) * (O_N / 16) / 8, TPB, 0, stream>>>(S, label, Mtri);
  k_gemm_G<<<(O_N / 16) * (O_N / 16) / 8, TPB, 0, stream>>>(W, Gc);
  k_gemm_T<<<(B_N / 16) * (O_N / 16) / 8, TPB, 0, stream>>>(S, label, Gc, T);
  k_recon<<<B_N, TPB, 0, stream>>>(S, X, T, label, recon);
  k_gemm_dw<<<(O_N / 16) * (D_N / 16) / 8, TPB, 0, stream>>>(Mtri, W, P, rinv, dw);
}